// SparseMemoryLayer_76596446756879
// MI455X (gfx1250) — compile-verified
//
#include <hip/hip_runtime.h>
#include <hip/hip_bf16.h>

// Shapes from the reference
#define HH   512
#define NN   16384
#define WW   64
#define KT   4
#define BB   64
#define DIN  512
#define ZDIM 2048   // 4*H
#define PDIM 194    // 3*W + 2

typedef float v2f __attribute__((ext_vector_type(2)));
typedef float v8f __attribute__((ext_vector_type(8)));

// ---------------------------------------------------------------------------
// WMMA fp32 16x16x4 fragment-loop: acc += A_tile(16xK) @ B_tile(Kx16)
// A already offset to tile row origin (row-major, leading dim lda)
// B already offset to tile col origin (row-major, leading dim ldb)
// Layouts per CDNA5 ISA 7.12.2:
//   A 16x4: lanes 0-15 M=0..15 {v0:K=0, v1:K=1}; lanes 16-31 {v0:K=2, v1:K=3}
//   B 4x16: lanes 0-15 N=0..15 {v0:K=0, v1:K=1}; lanes 16-31 {v0:K=2, v1:K=3}
// ---------------------------------------------------------------------------
__device__ __forceinline__ v8f wmma_frag_loop(v8f acc,
                                              const float* __restrict__ A, int lda,
                                              const float* __restrict__ Bm, int ldb,
                                              int K, int lane) {
    const int l  = lane & 15;
    const int k0 = (lane >> 4) << 1;   // 0 for lanes 0-15, 2 for lanes 16-31
    for (int k = 0; k < K; k += 4) {
        v2f a, b;
        a.x = A[l * lda + k + k0];
        a.y = A[l * lda + k + k0 + 1];
        b.x = Bm[(size_t)(k + k0) * ldb + l];
        b.y = Bm[(size_t)(k + k0 + 1) * ldb + l];
        acc = __builtin_amdgcn_wmma_f32_16x16x4_f32(false, a, false, b,
                                                    (short)0, acc, false, false);
    }
    return acc;
}

// ---------------------------------------------------------------------------
// K1: z = [inputs | r_prev] @ Wk + h_prev @ Uk + b_lstm   (64 x 2048)
// 512 tiles of 16x16, 8 waves/block -> 64 blocks of 256 threads
// ---------------------------------------------------------------------------
__global__ void k_zgemm(const float* __restrict__ inputs,
                        const float* __restrict__ r_prev,
                        const float* __restrict__ h_prev,
                        const float* __restrict__ Wk,
                        const float* __restrict__ Uk,
                        const float* __restrict__ b_lstm,
                        float* __restrict__ z) {
    const int wave = threadIdx.x >> 5, lane = threadIdx.x & 31;
    const int tile = blockIdx.x * 8 + wave;
    const int tm = (tile & 3) * 16;      // 4 M-tiles
    const int tn = (tile >> 2) * 16;     // 128 N-tiles
    v8f acc = {};
    acc = wmma_frag_loop(acc, inputs + tm * DIN, DIN, Wk + tn, ZDIM, DIN, lane);
    acc = wmma_frag_loop(acc, r_prev + tm * WW, WW,
                         Wk + (size_t)DIN * ZDIM + tn, ZDIM, WW, lane);
    acc = wmma_frag_loop(acc, h_prev + tm * HH, HH, Uk + tn, ZDIM, HH, lane);
    const int l = lane & 15, hi8 = (lane >> 4) * 8;
    for (int j = 0; j < 8; ++j) {
        const int row = tm + hi8 + j, col = tn + l;
        z[(size_t)row * ZDIM + col] = acc[j] + b_lstm[col];
    }
}

// ---------------------------------------------------------------------------
// K2: LSTM gates -> c_curr, h_curr
// ---------------------------------------------------------------------------
__device__ __forceinline__ float sigm(float x) { return 1.f / (1.f + __expf(-x)); }

__global__ void k_gates(const float* __restrict__ z,
                        const float* __restrict__ c_prev,
                        float* __restrict__ c_out,
                        float* __restrict__ h_out) {
    const int idx = blockIdx.x * blockDim.x + threadIdx.x;  // B*H = 32768
    const int b = idx >> 9, h = idx & 511;
    const float* zr = z + (size_t)b * ZDIM;
    const float i = sigm(zr[h]);
    const float f = sigm(zr[HH + h]);
    const float g = tanhf(zr[2 * HH + h]);
    const float o = sigm(zr[3 * HH + h]);
    const float c = f * c_prev[idx] + i * g;
    c_out[idx] = c;
    h_out[idx] = o * tanhf(c);
}

// ---------------------------------------------------------------------------
// K3: params = h_curr @ Wp + bp ; extract q_n, a, e, alpha, gamma
// one block per batch
// ---------------------------------------------------------------------------
__global__ void k_params(const float* __restrict__ h_curr,
                         const float* __restrict__ Wp,
                         const float* __restrict__ bp,
                         float* __restrict__ qn,
                         float* __restrict__ av,
                         float* __restrict__ ev,
                         float* __restrict__ alpha,
                         float* __restrict__ gamma) {
    __shared__ float p[PDIM];
    __shared__ float rn;
    const int b = blockIdx.x, tid = threadIdx.x;
    if (tid < PDIM) {
        const float* hr = h_curr + (size_t)b * HH;
        float acc = bp[tid];
        for (int k = 0; k < HH; ++k) acc += hr[k] * Wp[(size_t)k * PDIM + tid];
        p[tid] = acc;
    }
    __syncthreads();
    if (tid == 0) {
        float sq = 0.f;
        for (int w = 0; w < WW; ++w) sq += p[w] * p[w];
        rn = rsqrtf(fmaxf(sq, 1e-12f));
        alpha[b] = sigm(p[3 * WW]);
        gamma[b] = sigm(p[3 * WW + 1]);
    }
    __syncthreads();
    if (tid < WW) {
        qn[b * WW + tid] = p[tid] * rn;
        av[b * WW + tid] = p[WW + tid];
        ev[b * WW + tid] = sigm(p[2 * WW + tid]);
    }
}

// ---------------------------------------------------------------------------
// K4: sim[b,n] = q_n[b] . (M[b,n]/||M[b,n]||)   — 256MB streaming pass
// 8 threads per row, 32 rows per block; grid = B * N/32
// ---------------------------------------------------------------------------
__global__ void k_sim(const float* __restrict__ M,
                      const float* __restrict__ qn,
                      float* __restrict__ sim) {
    __shared__ float q[WW];
    const int b = blockIdx.x >> 9;          // / (N/32 = 512)
    const int nblk = blockIdx.x & 511;
    const int tid = threadIdx.x;
    if (tid < WW) q[tid] = qn[b * WW + tid];
    __syncthreads();
    const int rowi = tid >> 3, t8 = tid & 7;
    const int n = nblk * 32 + rowi;
    const float* mr = M + ((size_t)b * NN + n) * WW + t8 * 8;
    const float4 x0 = reinterpret_cast<const float4*>(mr)[0];
    const float4 x1 = reinterpret_cast<const float4*>(mr)[1];
    const float* qq = q + t8 * 8;
    float dot = x0.x * qq[0] + x0.y * qq[1] + x0.z * qq[2] + x0.w * qq[3]
              + x1.x * qq[4] + x1.y * qq[5] + x1.z * qq[6] + x1.w * qq[7];
    float sq = x0.x * x0.x + x0.y * x0.y + x0.z * x0.z + x0.w * x0.w
             + x1.x * x1.x + x1.y * x1.y + x1.z * x1.z + x1.w * x1.w;
    for (int off = 4; off >= 1; off >>= 1) {
        dot += __shfl_down(dot, off, 8);
        sq  += __shfl_down(sq,  off, 8);
    }
    if (t8 == 0) sim[(size_t)b * NN + n] = dot * rsqrtf(fmaxf(sq, 1e-12f));
}

// ---------------------------------------------------------------------------
// K5: per-batch top-4 -> softmax -> w_r, usage_curr, LRU argmax, r_curr
// one block (256 threads) per batch
// ---------------------------------------------------------------------------
__global__ void k_topk(const float* __restrict__ sim,
                       const float* __restrict__ usage_prev,
                       const float* __restrict__ M,
                       float* __restrict__ wr_out,
                       float* __restrict__ us_out,
                       float* __restrict__ r_out,
                       int* __restrict__ lru) {
    __shared__ float sv[1024];
    __shared__ int   si[1024];
    __shared__ float s_w[KT];
    __shared__ int   s_i[KT];
    __shared__ float ru[256];
    __shared__ int   rn[256];
    const int b = blockIdx.x, tid = threadIdx.x;
    const float* sb = sim + (size_t)b * NN;

    float tv[KT] = {-1e30f, -1e30f, -1e30f, -1e30f};
    int   ti[KT] = {-1, -1, -1, -1};
    for (int n = tid; n < NN; n += 256) {
        const float v = sb[n];
        if (v > tv[3]) {
            tv[3] = v; ti[3] = n;
            for (int k = 3; k > 0 && tv[k] > tv[k - 1]; --k) {
                float t = tv[k]; tv[k] = tv[k - 1]; tv[k - 1] = t;
                int u = ti[k]; ti[k] = ti[k - 1]; ti[k - 1] = u;
            }
        }
    }
    for (int k = 0; k < KT; ++k) { sv[tid * KT + k] = tv[k]; si[tid * KT + k] = ti[k]; }
    __syncthreads();
    if (tid == 0) {
        float bv[KT] = {-1e30f, -1e30f, -1e30f, -1e30f};
        int   bi[KT] = {-1, -1, -1, -1};
        for (int i = 0; i < 1024; ++i) {
            const float v = sv[i]; const int n = si[i];
            if (v > bv[3]) {
                bv[3] = v; bi[3] = n;
                for (int k = 3; k > 0 && bv[k] > bv[k - 1]; --k) {
                    float t = bv[k]; bv[k] = bv[k - 1]; bv[k - 1] = t;
                    int u = bi[k]; bi[k] = bi[k - 1]; bi[k - 1] = u;
                }
            }
        }
        const float m = bv[0];
        float s = 0.f, w[KT];
        for (int k = 0; k < KT; ++k) { w[k] = __expf(bv[k] - m); s += w[k]; }
        for (int k = 0; k < KT; ++k) { s_w[k] = w[k] / s; s_i[k] = bi[k]; }
    }
    __syncthreads();

    // dense usage_curr + w_r writes, local argmax of usage
    float bestu = -1e30f; int bestn = 0;
    for (int n = tid; n < NN; n += 256) {
        float wv = 0.f; bool hit = false;
        for (int k = 0; k < KT; ++k)
            if (s_i[k] == n) { wv = s_w[k]; hit = true; }
        wr_out[(size_t)b * NN + n] = wv;
        const float u = hit ? 0.f : (usage_prev[(size_t)b * NN + n] + 1.f);
        us_out[(size_t)b * NN + n] = u;
        if (u > bestu) { bestu = u; bestn = n; }
    }
    ru[tid] = bestu; rn[tid] = bestn;
    __syncthreads();
    if (tid == 0) {
        float bu = -1e30f; int bn = 0x7fffffff;
        for (int i = 0; i < 256; ++i)
            if (ru[i] > bu || (ru[i] == bu && rn[i] < bn)) { bu = ru[i]; bn = rn[i]; }
        lru[b] = bn;
    }
    // r_curr from the 4 selected rows
    if (tid < WW) {
        float r = 0.f;
        for (int k = 0; k < KT; ++k)
            r += s_w[k] * M[((size_t)b * NN + s_i[k]) * WW + tid];
        r_out[b * WW + tid] = r;
    }
}

// ---------------------------------------------------------------------------
// K6: M_curr = M_prev * (1 - ww*e) + ww*a   — 512MB streaming pass
// ww[b,n] = alpha*(gamma*wr_prev + (1-gamma)*delta(n,lru)) computed on the fly
// one float4 per thread; 1024 blocks per batch
// ---------------------------------------------------------------------------
__global__ void k_mupdate(const float* __restrict__ M,
                          const float* __restrict__ wr_prev,
                          const float* __restrict__ ev,
                          const float* __restrict__ av,
                          const float* __restrict__ alpha,
                          const float* __restrict__ gamma,
                          const int* __restrict__ lru,
                          float* __restrict__ Mout) {
    __shared__ float se[WW];
    __shared__ float sa[WW];
    const int b = blockIdx.x >> 10;
    const int blk = blockIdx.x & 1023;
    const int tid = threadIdx.x;
    if (tid < WW) se[tid] = ev[b * WW + tid];
    else if (tid < 2 * WW) sa[tid - WW] = av[b * WW + tid - WW];
    __syncthreads();
    const float al = alpha[b], ga = gamma[b];
    const int lr = lru[b];
    const int ei = blk * 256 + tid;        // float4 index within batch
    const int n = ei >> 4, w4 = ei & 15;
    const float wrp = wr_prev[(size_t)b * NN + n];
    const float ww = al * (ga * wrp + (1.f - ga) * ((n == lr) ? 1.f : 0.f));
    const size_t off = ((size_t)b * NN + n) * WW + w4 * 4;
    const float4 m = *reinterpret_cast<const float4*>(M + off);
    const float* e4 = se + w4 * 4;
    const float* a4 = sa + w4 * 4;
    float4 o;
    o.x = m.x * (1.f - ww * e4[0]) + ww * a4[0];
    o.y = m.y * (1.f - ww * e4[1]) + ww * a4[1];
    o.z = m.z * (1.f - ww * e4[2]) + ww * a4[2];
    o.w = m.w * (1.f - ww * e4[3]) + ww * a4[3];
    *reinterpret_cast<float4*>(Mout + off) = o;
}

// ---------------------------------------------------------------------------
// K7: y_out = [h_curr | r_curr] @ Wf + bf   (64 x 512) — WMMA
// 128 tiles, 8 waves/block -> 16 blocks
// ---------------------------------------------------------------------------
__global__ void k_ygemm(const float* __restrict__ h_curr,
                        const float* __restrict__ r_curr,
                        const float* __restrict__ Wf,
                        const float* __restrict__ bf,
                        float* __restrict__ y) {
    const int wave = threadIdx.x >> 5, lane = threadIdx.x & 31;
    const int tile = blockIdx.x * 8 + wave;
    const int tm = (tile & 3) * 16;      // 4 M-tiles
    const int tn = (tile >> 2) * 16;     // 32 N-tiles
    v8f acc = {};
    acc = wmma_frag_loop(acc, h_curr + tm * HH, HH, Wf + tn, HH, HH, lane);
    acc = wmma_frag_loop(acc, r_curr + tm * WW, WW,
                         Wf + (size_t)HH * HH + tn, HH, WW, lane);
    const int l = lane & 15, hi8 = (lane >> 4) * 8;
    for (int j = 0; j < 8; ++j) {
        const int row = tm + hi8 + j, col = tn + l;
        y[(size_t)row * HH + col] = acc[j] + bf[col];
    }
}

// ---------------------------------------------------------------------------
// launch
// ---------------------------------------------------------------------------
extern "C" void kernel_launch(void* const* d_in, const int* in_sizes, int n_in,
                              void* d_out, int out_size, void* d_ws, size_t ws_size,
                              hipStream_t stream) {
    (void)in_sizes; (void)n_in; (void)out_size; (void)ws_size;
    const float* inputs     = (const float*)d_in[0];
    const float* h_prev     = (const float*)d_in[1];
    const float* c_prev     = (const float*)d_in[2];
    const float* M_prev     = (const float*)d_in[3];
    const float* wr_prev    = (const float*)d_in[4];
    const float* usage_prev = (const float*)d_in[5];
    const float* r_prev     = (const float*)d_in[6];
    const float* Wk         = (const float*)d_in[7];
    const float* Uk         = (const float*)d_in[8];
    const float* b_lstm     = (const float*)d_in[9];
    const float* Wp         = (const float*)d_in[10];
    const float* bp         = (const float*)d_in[11];
    const float* Wf         = (const float*)d_in[12];
    const float* bf         = (const float*)d_in[13];

    float* out = (float*)d_out;
    // output layout: y, h, c, M, w_r, usage, r
    const size_t o_y = 0;
    const size_t o_h = o_y + (size_t)BB * HH;          // 32768
    const size_t o_c = o_h + (size_t)BB * HH;          // 65536
    const size_t o_M = o_c + (size_t)BB * HH;          // 98304
    const size_t o_wr = o_M + (size_t)BB * NN * WW;    // 67207168
    const size_t o_us = o_wr + (size_t)BB * NN;        // 68255744
    const size_t o_r = o_us + (size_t)BB * NN;         // 69304320

    float* ws = (float*)d_ws;
    float* ws_z   = ws;                 // 131072 (B*4H)
    float* ws_qn  = ws + 131072;        // 4096
    float* ws_a   = ws + 135168;        // 4096
    float* ws_e   = ws + 139264;        // 4096
    float* ws_al  = ws + 143360;        // 64
    float* ws_ga  = ws + 143424;        // 64
    float* ws_sim = ws + 143488;        // 1048576
    int*   ws_lru = (int*)(ws + 1192064); // 64

    k_zgemm<<<64, 256, 0, stream>>>(inputs, r_prev, h_prev, Wk, Uk, b_lstm, ws_z);
    k_gates<<<128, 256, 0, stream>>>(ws_z, c_prev, out + o_c, out + o_h);
    k_params<<<BB, 256, 0, stream>>>(out + o_h, Wp, bp, ws_qn, ws_a, ws_e, ws_al, ws_ga);
    k_sim<<<BB * (NN / 32), 256, 0, stream>>>(M_prev, ws_qn, ws_sim);
    k_topk<<<BB, 256, 0, stream>>>(ws_sim, usage_prev, M_prev,
                                   out + o_wr, out + o_us, out + o_r, ws_lru);
    k_mupdate<<<BB * 1024, 256, 0, stream>>>(M_prev, wr_prev, ws_e, ws_a,
                                             ws_al, ws_ga, ws_lru, out + o_M);
    k_ygemm<<<16, 256, 0, stream>>>(out + o_h, out + o_r, Wf, bf, out + o_y);
}